// Reservoir_9371618640550
// MI455X (gfx1250) — compile-verified
//
#include <hip/hip_runtime.h>
#include <cmath>

typedef float v2f __attribute__((ext_vector_type(2)));
typedef float v8f __attribute__((ext_vector_type(8)));

static constexpr int kSeq    = 4096;
static constexpr int kIn     = 128;
static constexpr int kRes    = 2048;
static constexpr int kNWG    = 64;               // workgroups in the scan (1 per WGP, 256KB+ LDS each)
static constexpr int kRows   = kRes / kNWG;      // 32 rows of W_res per workgroup
static constexpr int kStride = kRes + 16;        // LDS row stride (mod 64 == 16 -> conflict-free teams)
static constexpr int kBlk    = 256;
static constexpr size_t kSmemBytes = (size_t)(kRows * kStride + kRes) * sizeof(float); // 272384 B

// ---------------------------------------------------------------------------
// Kernel 1: u_proj = input_data @ W_in^T  via V_WMMA_F32_16X16X4_F32
// M=4096, N=2048, K=128. One 16x16 tile per wave, 32 WMMA ops per tile.
// Also zeroes the grid-barrier counter for the scan kernel (stream-ordered).
// ---------------------------------------------------------------------------
__global__ __launch_bounds__(kBlk) void uproj_gemm_wmma(
    const float* __restrict__ X,    // (4096,128)
    const float* __restrict__ Win,  // (2048,128)
    float* __restrict__ U,          // (4096,2048) out
    unsigned* __restrict__ counter)
{
    if (blockIdx.x == 0 && threadIdx.x == 0) {
        __hip_atomic_store(counter, 0u, __ATOMIC_RELAXED, __HIP_MEMORY_SCOPE_AGENT);
    }
    const int wave = threadIdx.x >> 5;
    const int lane = threadIdx.x & 31;
    const int half = lane >> 4;     // 0: lanes 0-15, 1: lanes 16-31
    const int lm   = lane & 15;

    const int tile = blockIdx.x * (kBlk / 32) + wave;   // 32768 tiles total
    const int tm   = tile >> 7;                         // 2048/16 = 128 N-tiles
    const int tn   = tile & 127;

    const float* arow = X   + (size_t)(tm * 16 + lm) * kIn;  // A: 16x4 f32 fragment rows
    const float* brow = Win + (size_t)(tn * 16 + lm) * kIn;  // B = Win^T: B[k][n] = Win[n][k]

    v8f c = {0.f, 0.f, 0.f, 0.f, 0.f, 0.f, 0.f, 0.f};
#pragma unroll
    for (int k = 0; k < kIn; k += 4) {
        // ISA layout: lane l holds A[l%16][k + 2*(l/16) + {0,1}], symmetric for B.
        v2f a = *(const v2f*)(arow + k + 2 * half);
        v2f b = *(const v2f*)(brow + k + 2 * half);
        c = __builtin_amdgcn_wmma_f32_16x16x4_f32(false, a, false, b, (short)0, c, false, false);
    }
    // D layout: VGPR v -> row M = v + 8*half, col N = lm. INPUT_SCALE == 1.0.
    float* orow = U + (size_t)(tm * 16 + 8 * half) * kRes + tn * 16 + lm;
#pragma unroll
    for (int v = 0; v < 8; ++v) orow[(size_t)v * kRes] = c[v];
}

// ---------------------------------------------------------------------------
// Kernel 2: persistent reservoir scan. 64 workgroups, each caches a 32-row
// slice of W_res in LDS (264 KB) + the full 8 KB state vector. Per step:
// 8-lane teams compute one matvec row each from LDS, tanh+scale, write the
// new state into d_out, monotonic global-atomic grid barrier, then all WGs
// pull the fresh state straight into LDS with GLOBAL_LOAD_ASYNC_TO_LDS_B128
// (ASYNCcnt-tracked, no VGPR round-trip).
// ---------------------------------------------------------------------------
__global__ __launch_bounds__(kBlk) void reservoir_scan(
    const float* __restrict__ Wres,  // (2048,2048)
    const float* __restrict__ s0,    // (2048,)
    const float* __restrict__ U,     // (4096,2048) u_proj
    float* __restrict__ out,         // (4097,2048) states
    unsigned* __restrict__ counter)
{
    extern __shared__ float smem[];
    float* sW = smem;                       // kRows x kStride (padded)
    float* sS = smem + kRows * kStride;     // kRes

    const int tid  = threadIdx.x;
    const int row0 = blockIdx.x * kRows;

    // Cache this WG's W_res slice into LDS (vectorized, coalesced).
    for (int idx = tid * 4; idx < kRows * kRes; idx += kBlk * 4) {
        const int r = idx / kRes, cc = idx % kRes;
        const float4 w = *(const float4*)(Wres + (size_t)(row0 + r) * kRes + cc);
        *(float4*)(sW + r * kStride + cc) = w;
    }
    // Initial state -> LDS; also emit output row 0 (this WG's slice).
    for (int i = tid; i < kRes; i += kBlk) sS[i] = s0[i];
    for (int i = tid; i < kRows; i += kBlk) out[row0 + i] = s0[row0 + i];
    __syncthreads();

    const int team = tid >> 3;   // 32 teams = 32 rows
    const int tl   = tid & 7;    // 8 lanes per team
    const float* wrow = sW + team * kStride;
    const float inv_sqrt_n = 0.022097086912079608f;  // 1/sqrt(2048)

    // Per-thread async-copy addresses: 256 threads x 16B = 4KB per issue;
    // two issues (offset 0 / 4096) cover the full 8KB state vector.
    // Flat->LDS mapping keeps the LDS byte address in the low 32 bits.
    const unsigned lds_s = (unsigned)(uintptr_t)sS + (unsigned)(tid * 16);

    for (int t = 0; t < kSeq; ++t) {
        // Prefetch u_proj for this row (overlaps with the dot product).
        float u = 0.f;
        if (tl == 0) u = U[(size_t)t * kRes + row0 + team];

        float acc = 0.f;
#pragma unroll 8
        for (int m = 0; m < kRes / 32; ++m) {          // 64 float4 chunks/thread
            const int cc = (tl + 8 * m) * 4;
            const float4 w = *(const float4*)(wrow + cc);
            const float4 s = *(const float4*)(sS + cc);
            acc += w.x * s.x + w.y * s.y + w.z * s.z + w.w * s.w;
        }
        // 8-lane tree reduction (wave32-safe: stays inside 8-lane groups).
        acc += __shfl_xor(acc, 4, 32);
        acc += __shfl_xor(acc, 2, 32);
        acc += __shfl_xor(acc, 1, 32);

        if (tl == 0) {
            const float v = tanhf(u + 0.9f * acc) * inv_sqrt_n;
            out[(size_t)(t + 1) * kRes + row0 + team] = v;
        }
        __threadfence();          // make state stores agent-visible
        __syncthreads();

        if (tid == 0) {
            __hip_atomic_fetch_add(counter, 1u, __ATOMIC_RELEASE, __HIP_MEMORY_SCOPE_AGENT);
            const unsigned target = (unsigned)kNWG * (unsigned)(t + 1);
            while (__hip_atomic_load(counter, __ATOMIC_ACQUIRE, __HIP_MEMORY_SCOPE_AGENT) < target) {
                __builtin_amdgcn_s_sleep(1);
            }
        }
        __syncthreads();
        __threadfence();          // invalidate stale cached lines before reload

        // Async-copy the complete new state vector directly into LDS.
        // INST_OFFSET is added to BOTH the LDS and global addresses (ISA 10.x).
        {
            const unsigned long long gaddr =
                (unsigned long long)(uintptr_t)(out + (size_t)(t + 1) * kRes) +
                (unsigned long long)(tid * 16);
            asm volatile("global_load_async_to_lds_b128 %0, %1, off"
                         :: "v"(lds_s), "v"(gaddr) : "memory");
            asm volatile("global_load_async_to_lds_b128 %0, %1, off offset:4096"
                         :: "v"(lds_s), "v"(gaddr) : "memory");
            asm volatile("s_wait_asynccnt 0x0" ::: "memory");
        }
        __syncthreads();
    }
}

// ---------------------------------------------------------------------------
extern "C" void kernel_launch(void* const* d_in, const int* in_sizes, int n_in,
                              void* d_out, int out_size, void* d_ws, size_t ws_size,
                              hipStream_t stream) {
    (void)in_sizes; (void)n_in; (void)out_size; (void)ws_size;
    const float* X    = (const float*)d_in[0];   // (4096,128)
    const float* s0   = (const float*)d_in[1];   // (2048,)
    const float* Win  = (const float*)d_in[2];   // (2048,128)
    const float* Wres = (const float*)d_in[3];   // (2048,2048)
    float* out = (float*)d_out;                  // (4097,2048)

    float*    U       = (float*)d_ws;                                    // 32 MB u_proj
    unsigned* counter = (unsigned*)((char*)d_ws + (size_t)kSeq * kRes * sizeof(float));

    // Allow 272 KB dynamic LDS (CDNA5 WGP has 320 KB).
    (void)hipFuncSetAttribute((const void*)reservoir_scan,
                              hipFuncAttributeMaxDynamicSharedMemorySize,
                              (int)kSmemBytes);

    // GEMM: 256x128 tiles of 16x16, 8 waves/block -> 4096 blocks. Also zeroes counter.
    const int tiles  = (4096 / 16) * (kRes / 16);
    const int blocks = tiles / (kBlk / 32);
    uproj_gemm_wmma<<<blocks, kBlk, 0, stream>>>(X, Win, U, counter);

    // Persistent scan: 64 WGs, one per WGP (256KB+ LDS each).
    reservoir_scan<<<kNWG, kBlk, kSmemBytes, stream>>>(Wres, s0, U, out, counter);
}